// Net_2_36155034697768
// MI455X (gfx1250) — compile-verified
//
#include <hip/hip_runtime.h>
#include <math.h>

typedef __attribute__((ext_vector_type(2))) float v2f;
typedef __attribute__((ext_vector_type(8))) float v8f;

#define N_CELLS 20
#define IN 8
#define HID 10
#define FEAT 200
#define H1 50
#define H2 10

__device__ __forceinline__ float sigmoidf_(float v) {
    return 1.0f / (1.0f + expf(-v));
}

__global__ __launch_bounds__(128) void lstm_decoder_fused(
    const float* __restrict__ x,
    const float* __restrict__ h0,
    const float* __restrict__ c0,
    const float* __restrict__ W_ih,
    const float* __restrict__ W_hh,
    const float* __restrict__ b_ih,
    const float* __restrict__ b_hh,
    const float* __restrict__ W1,
    const float* __restrict__ b1,
    const float* __restrict__ W2,
    const float* __restrict__ b2,
    const float* __restrict__ W3,
    const float* __restrict__ b3,
    float* __restrict__ out)
{
    __shared__ float feat[FEAT];   // decoder input, (gen, opp) order
    __shared__ float h1s[64];      // padded 50 -> 64 (rows 50..63 are garbage, never read)
    __shared__ float h2s[H2];

    const int tid = threadIdx.x;

    // Prefetch W1 (40 KB) toward cache while phase A computes (global_prefetch_b8).
    for (int off = tid * 64; off < H1 * FEAT; off += 128 * 64)
        __builtin_prefetch(W1 + off, 0, 1);

    // ---- Phase A: one LSTM step for 20 cells x 10 hidden units ----
    for (int t = tid; t < N_CELLS * HID; t += 128) {
        const int k = t / HID;
        const int j = t - k * HID;
        const float* wih = W_ih + (k * 4 * HID) * IN;
        const float* whh = W_hh + (k * 4 * HID) * HID;
        const int gb = k * 4 * HID;
        float gi = b_ih[gb + j]           + b_hh[gb + j];
        float gf = b_ih[gb + HID + j]     + b_hh[gb + HID + j];
        float gg = b_ih[gb + 2 * HID + j] + b_hh[gb + 2 * HID + j];
        float go = b_ih[gb + 3 * HID + j] + b_hh[gb + 3 * HID + j];
        #pragma unroll
        for (int i = 0; i < IN; ++i) {
            const float xv = x[i];
            gi += wih[(j) * IN + i] * xv;
            gf += wih[(HID + j) * IN + i] * xv;
            gg += wih[(2 * HID + j) * IN + i] * xv;
            go += wih[(3 * HID + j) * IN + i] * xv;
        }
        #pragma unroll
        for (int hh = 0; hh < HID; ++hh) {
            const float hv = h0[k * HID + hh];
            gi += whh[(j) * HID + hh] * hv;
            gf += whh[(HID + j) * HID + hh] * hv;
            gg += whh[(2 * HID + j) * HID + hh] * hv;
            go += whh[(3 * HID + j) * HID + hh] * hv;
        }
        const float ig = sigmoidf_(gi);
        const float fg = sigmoidf_(gf);
        const float g2 = tanhf(gg);
        const float og = sigmoidf_(go);
        const float c  = fg * c0[k * HID + j] + ig * g2;
        const float h  = og * tanhf(c);
        // cells 10..19 (gen) first, then 0..9 (opp)
        const int fidx = (k >= 10) ? ((k - 10) * HID + j) : (100 + k * HID + j);
        feat[fidx] = h;
    }
    __syncthreads();

    // ---- Phase B: h1 = tanh(W1 @ feat + b1) on V_WMMA_F32_16X16X4_F32 ----
    // 4 waves, one 16-row M-tile each (M padded 50->64); K = 200 in 50 steps of 4.
    // No guards needed:
    //  * OOB rows are CLAMPED to row 0 (memory-safe); their D rows are never read.
    //  * Every lane loads the same feat pair for B -> all 16 D columns hold the
    //    answer; we read column 0. Unconditional global_load_b64 + ds_load_b64.
    {
        const int wave  = tid >> 5;
        const int lane  = tid & 31;
        const int mrow  = wave * 16 + (lane & 15);  // 0..63, valid < 50
        const int khalf = lane >> 4;                // 0 or 1
        const int crow  = (mrow < H1) ? mrow : 0;   // clamp for memory safety
        const float* Arow = W1 + crow * FEAT + 2 * khalf;

        v8f acc0 = {0.f, 0.f, 0.f, 0.f, 0.f, 0.f, 0.f, 0.f};
        v8f acc1 = {0.f, 0.f, 0.f, 0.f, 0.f, 0.f, 0.f, 0.f};
        #pragma unroll
        for (int k0 = 0; k0 < FEAT; k0 += 8) {
            {
                const v2f a = *(const v2f*)(Arow + k0);
                const v2f b = *(const v2f*)(&feat[k0 + 2 * khalf]);
                acc0 = __builtin_amdgcn_wmma_f32_16x16x4_f32(
                    false, a, false, b, (short)0, acc0, false, false);
            }
            {
                const v2f a = *(const v2f*)(Arow + k0 + 4);
                const v2f b = *(const v2f*)(&feat[k0 + 4 + 2 * khalf]);
                acc1 = __builtin_amdgcn_wmma_f32_16x16x4_f32(
                    false, a, false, b, (short)0, acc1, false, false);
            }
        }
        // D column 0: lane 0 holds rows m0..m0+7 in acc[0..7], lane 16 rows m0+8..m0+15.
        if ((lane & 15) == 0) {
            #pragma unroll
            for (int r = 0; r < 8; ++r) {
                const int m  = wave * 16 + khalf * 8 + r;      // 0..63
                const int mc = (m < H1) ? m : 0;               // clamp b1 read
                h1s[m] = tanhf(acc0[r] + acc1[r] + b1[mc]);    // rows >=50: garbage, unread
            }
        }
    }
    __syncthreads();

    // ---- Phase C: h2 = tanh(W2 @ h1 + b2) ----
    if (tid < H2) {
        float s = b2[tid];
        #pragma unroll 5
        for (int i = 0; i < H1; ++i) s += W2[tid * H1 + i] * h1s[i];
        h2s[tid] = tanhf(s);
    }
    __syncthreads();

    // ---- Phase D: out = tanh(W3 @ h2 + b3) ----
    if (tid == 0) {
        float s = b3[0];
        #pragma unroll
        for (int i = 0; i < H2; ++i) s += W3[i] * h2s[i];
        out[0] = tanhf(s);
    }
}

extern "C" void kernel_launch(void* const* d_in, const int* in_sizes, int n_in,
                              void* d_out, int out_size, void* d_ws, size_t ws_size,
                              hipStream_t stream) {
    (void)in_sizes; (void)n_in; (void)out_size; (void)d_ws; (void)ws_size;
    const float* x    = (const float*)d_in[0];
    const float* h0   = (const float*)d_in[1];
    const float* c0   = (const float*)d_in[2];
    const float* W_ih = (const float*)d_in[3];
    const float* W_hh = (const float*)d_in[4];
    const float* b_ih = (const float*)d_in[5];
    const float* b_hh = (const float*)d_in[6];
    const float* W1   = (const float*)d_in[7];
    const float* b1   = (const float*)d_in[8];
    const float* W2   = (const float*)d_in[9];
    const float* b2   = (const float*)d_in[10];
    const float* W3   = (const float*)d_in[11];
    const float* b3   = (const float*)d_in[12];
    float* out = (float*)d_out;

    lstm_decoder_fused<<<1, 128, 0, stream>>>(
        x, h0, c0, W_ih, W_hh, b_ih, b_hh, W1, b1, W2, b2, W3, b3, out);
}